// VarSelfAttention_86612310491324
// MI455X (gfx1250) — compile-verified
//
#include <hip/hip_runtime.h>
#include <math.h>

typedef __attribute__((ext_vector_type(2))) float v2f;
typedef __attribute__((ext_vector_type(8))) float v8f;

#define B_   4
#define S_   2048
#define D_   1024
#define H_   16
#define DH_  64
#define PSTR 2052   // padded LDS row stride: bank = (4*row + col) % 64 -> conflict-free

#define WMMA_F32(A, Bv, C) \
    __builtin_amdgcn_wmma_f32_16x16x4_f32(false, (A), false, (Bv), (short)0, (C), false, false)

// ---------------------------------------------------------------------------
// Kernel 1: fused q/k projection  qk = x @ W^T + b  ->  Q[B,H,S,dh], K[B,H,S,dh]
// Each wave: 32x64 tile (8 accumulators; 2 A-frags reuse 4 B-frags).
// Register double-buffered K-loop so loads for step k+4 are in flight while
// WMMAs consume step k (partial s_wait_loadcnt instead of full drains).
// ---------------------------------------------------------------------------
__global__ __launch_bounds__(128) void qk_proj_kernel(
    const float* __restrict__ x,      // [B*S, D]
    const float* __restrict__ W,      // [2D, D] row-major
    const float* __restrict__ bias,   // [2D]
    float* __restrict__ Qws,          // [B,H,S,dh]
    float* __restrict__ Kws)          // [B,H,S,dh]
{
    const int lane = threadIdx.x & 31;
    const int wave = threadIdx.x >> 5;
    const int lr   = lane & 15;
    const int hi   = lane >> 4;

    const int m0 = blockIdx.x * 32;
    const int n0 = blockIdx.y * 256 + wave * 64;

    v8f c0[4] = { {}, {}, {}, {} };   // rows m0 .. m0+15
    v8f c1[4] = { {}, {}, {}, {} };   // rows m0+16 .. m0+31

    const float* xr0 = x + (size_t)(m0 + lr) * D_ + hi * 2;
    const float* xr1 = xr0 + (size_t)16 * D_;
    const float* w0  = W + (size_t)(n0 +  0 + lr) * D_ + hi * 2;
    const float* w1  = W + (size_t)(n0 + 16 + lr) * D_ + hi * 2;
    const float* w2  = W + (size_t)(n0 + 32 + lr) * D_ + hi * 2;
    const float* w3  = W + (size_t)(n0 + 48 + lr) * D_ + hi * 2;

    // prologue: fragments for k0 = 0
    v2f a0c = *(const v2f*)(xr0), a1c = *(const v2f*)(xr1);
    v2f b0c = *(const v2f*)(w0),  b1c = *(const v2f*)(w1);
    v2f b2c = *(const v2f*)(w2),  b3c = *(const v2f*)(w3);

    for (int k0 = 0; k0 < D_ - 4; k0 += 4) {
        const int kn = k0 + 4;
        // issue next step's loads BEFORE consuming current fragments
        v2f a0n = *(const v2f*)(xr0 + kn), a1n = *(const v2f*)(xr1 + kn);
        v2f b0n = *(const v2f*)(w0 + kn),  b1n = *(const v2f*)(w1 + kn);
        v2f b2n = *(const v2f*)(w2 + kn),  b3n = *(const v2f*)(w3 + kn);

        c0[0] = WMMA_F32(a0c, b0c, c0[0]);  c1[0] = WMMA_F32(a1c, b0c, c1[0]);
        c0[1] = WMMA_F32(a0c, b1c, c0[1]);  c1[1] = WMMA_F32(a1c, b1c, c1[1]);
        c0[2] = WMMA_F32(a0c, b2c, c0[2]);  c1[2] = WMMA_F32(a1c, b2c, c1[2]);
        c0[3] = WMMA_F32(a0c, b3c, c0[3]);  c1[3] = WMMA_F32(a1c, b3c, c1[3]);

        a0c = a0n; a1c = a1n; b0c = b0n; b1c = b1n; b2c = b2n; b3c = b3n;
    }
    // epilogue
    c0[0] = WMMA_F32(a0c, b0c, c0[0]);  c1[0] = WMMA_F32(a1c, b0c, c1[0]);
    c0[1] = WMMA_F32(a0c, b1c, c0[1]);  c1[1] = WMMA_F32(a1c, b1c, c1[1]);
    c0[2] = WMMA_F32(a0c, b2c, c0[2]);  c1[2] = WMMA_F32(a1c, b2c, c1[2]);
    c0[3] = WMMA_F32(a0c, b3c, c0[3]);  c1[3] = WMMA_F32(a1c, b3c, c1[3]);

    // store with q/k de-interleave.  C layout: VGPR r -> M = r + 8*hi, N = lr
    #pragma unroll
    for (int blk = 0; blk < 2; ++blk) {
        #pragma unroll
        for (int r = 0; r < 8; ++r) {
            const int m  = m0 + blk * 16 + r + hi * 8;
            const int bb = m >> 11;              // / S_
            const int s  = m & (S_ - 1);
            #pragma unroll
            for (int j = 0; j < 4; ++j) {
                const float cv = blk ? c1[j][r] : c0[j][r];
                const int f = n0 + j * 16 + lr;
                const float val = cv + bias[f];
                const int h = f >> 7;
                const int d = (f >> 1) & (DH_ - 1);
                float* dst = (f & 1) ? Kws : Qws;
                dst[(((size_t)bb * H_ + h) * S_ + s) * DH_ + d] = val;
            }
        }
    }
}

// --------------------------- attention helpers -----------------------------
__device__ __forceinline__ void load_kfrag(const float* kbase, int jt, int lr, int hi,
                                           v2f (&dst)[16]) {
    const float* krow = kbase + (size_t)(jt * 16 + lr) * DH_ + hi * 2;
    #pragma unroll
    for (int ks = 0; ks < 16; ++ks) dst[ks] = *(const v2f*)(krow + ks * 4);
}

__device__ __forceinline__ void score_store(float* P, const v2f (&aq)[16],
                                            const v2f (&kb)[16], int jt, int lr, int hi) {
    v8f cs = {};
    #pragma unroll
    for (int ks = 0; ks < 16; ++ks) cs = WMMA_F32(aq[ks], kb[ks], cs);
    #pragma unroll
    for (int r = 0; r < 8; ++r)
        P[(r + hi * 8) * PSTR + jt * 16 + lr] = cs[r] * 0.125f;     // dh^-0.5
}

__device__ __forceinline__ void load_vfrag(const float* vbase, int k, int lr, int hi,
                                           v2f (&dst)[4]) {
    const float* vp = vbase + (size_t)(k + hi * 2) * D_;
    #pragma unroll
    for (int j = 0; j < 4; ++j) {
        const int d = j * 16 + lr;
        v2f t = { vp[d], vp[D_ + d] };
        dst[j] = t;
    }
}

__device__ __forceinline__ void pv_step(const float* P, int k, int lr, int hi,
                                        const v2f (&bf)[4], v8f (&c4)[4]) {
    v2f a = *(const v2f*)(P + lr * PSTR + k + hi * 2);              // ds_load_b64
    #pragma unroll
    for (int j = 0; j < 4; ++j) c4[j] = WMMA_F32(a, bf[j], c4[j]);
}

// ---------------------------------------------------------------------------
// Kernel 2: fused attention for one (b,h) and one 16-query strip.
// 4 waves/block share the LDS score strip:
//   phase 1: waves split jt tiles of sim = QK^T/8        (pipelined K loads)
//   phase 2: 128 threads row-mean reduction
//   phase 3: SELU(sim - mean) in LDS
//   phase 4: waves split the k range of P@V, ds_add_f32 LDS reduction
// ---------------------------------------------------------------------------
__global__ __launch_bounds__(128) void attn_kernel(
    const float* __restrict__ x,      // [B,S,D]  (V source, identity proj)
    const float* __restrict__ Qws,    // [B,H,S,dh]
    const float* __restrict__ Kws,    // [B,H,S,dh]
    float* __restrict__ out)          // [B,S,D]
{
    extern __shared__ float smem[];
    float* P       = smem;                    // [16][PSTR]
    float* meanbuf = P + 16 * PSTR;           // [16]
    float* rs      = meanbuf + 16;            // [16][8] rowsum partials
    float* acc     = rs + 16 * 8;             // [16][64] PV reduction

    const int t    = threadIdx.x;
    const int lane = t & 31;
    const int wave = t >> 5;                  // 0..3
    const int lr   = lane & 15;
    const int hi   = lane >> 4;

    const int bh = blockIdx.x;
    const int b  = bh >> 4;
    const int h  = bh & (H_ - 1);
    const int i0 = blockIdx.y * 16;

    const float* qbase = Qws + ((size_t)bh * S_) * DH_;
    const float* kbase = Kws + ((size_t)bh * S_) * DH_;

    // every wave holds the full 16x64 Q strip as A-fragments (32 VGPRs)
    v2f aq[16];
    {
        const float* qrow = qbase + (size_t)(i0 + lr) * DH_ + hi * 2;
        #pragma unroll
        for (int ks = 0; ks < 16; ++ks) aq[ks] = *(const v2f*)(qrow + ks * 4);
    }

    // ---- Phase 1: scores -> LDS; wave w owns jt = w, w+4, w+8, ... --------
    {
        v2f kA[16], kB[16];
        load_kfrag(kbase, wave, lr, hi, kA);
        for (int jt = wave; jt < S_ / 16; jt += 8) {
            if (jt + 16 < S_ / 16)
                __builtin_prefetch(kbase + (size_t)((jt + 16) * 16 + lr) * DH_, 0, 1);
            load_kfrag(kbase, jt + 4, lr, hi, kB);     // in flight during kA WMMAs
            score_store(P, aq, kA, jt, lr, hi);
            if (jt + 8 < S_ / 16) load_kfrag(kbase, jt + 8, lr, hi, kA);
            score_store(P, aq, kB, jt + 4, lr, hi);
        }
    }
    __syncthreads();

    // ---- Phase 2: row means (thread t -> row t%16, column segment t/16) ---
    {
        const int row = t & 15, seg = t >> 4;          // seg 0..7
        float partial = 0.0f;
        const float* prow = P + row * PSTR;
        for (int cc = seg; cc < S_; cc += 8) partial += prow[cc];
        rs[row * 8 + seg] = partial;
    }
    __syncthreads();
    if (t < 16) {
        float s = 0.0f;
        #pragma unroll
        for (int i = 0; i < 8; ++i) s += rs[t * 8 + i];
        meanbuf[t] = s * (1.0f / (float)S_);
    }
    __syncthreads();

    // ---- Phase 3: SELU(sim - mean) in place -------------------------------
    {
        const float kScale = 1.0507009873554805f;
        const float kAlpha = 1.6732632423543772f;
        for (int idx = t; idx < 16 * S_; idx += 128) {
            const int row = idx >> 11;
            const int col = idx & (S_ - 1);
            const float v = P[row * PSTR + col] - meanbuf[row];
            P[row * PSTR + col] = (v > 0.0f) ? kScale * v
                                             : kScale * kAlpha * (__expf(v) - 1.0f);
        }
        for (int idx = t; idx < 16 * DH_; idx += 128) acc[idx] = 0.0f;
    }
    __syncthreads();

    // ---- Phase 4: out = P @ V; wave w owns k in [w*512, w*512+512) --------
    {
        v8f c4[4] = { {}, {}, {}, {} };
        const float* vbase = x + ((size_t)b * S_) * D_ + h * DH_;
        const int kstart = wave * (S_ / 4);
        const int kend   = kstart + S_ / 4;

        v2f bfA[4], bfB[4];
        load_vfrag(vbase, kstart, lr, hi, bfA);
        for (int k = kstart; k < kend; k += 8) {
            load_vfrag(vbase, k + 4, lr, hi, bfB);     // in flight during bfA WMMAs
            pv_step(P, k, lr, hi, bfA, c4);
            if (k + 8 < kend) load_vfrag(vbase, k + 8, lr, hi, bfA);
            pv_step(P, k + 4, lr, hi, bfB, c4);
        }

        // cross-wave reduction in LDS (ds_add_f32)
        #pragma unroll
        for (int r = 0; r < 8; ++r) {
            const int m = r + hi * 8;
            #pragma unroll
            for (int j = 0; j < 4; ++j)
                atomicAdd(&acc[m * DH_ + j * 16 + lr], c4[j][r]);
        }
    }
    __syncthreads();

    // ---- write out (exclusive region per block) ---------------------------
    {
        const float oscale = 0.022097086912079608f;    // S^-0.5
        float* obase = out + ((size_t)b * S_ + i0) * D_ + h * DH_;
        for (int idx = t; idx < 16 * DH_; idx += 128) {
            const int m = idx >> 6;
            const int d = idx & (DH_ - 1);
            obase[(size_t)m * D_ + d] = acc[idx] * oscale;
        }
    }
}

// ---------------------------------------------------------------------------
extern "C" void kernel_launch(void* const* d_in, const int* in_sizes, int n_in,
                              void* d_out, int out_size, void* d_ws, size_t ws_size,
                              hipStream_t stream) {
    (void)in_sizes; (void)n_in; (void)out_size; (void)ws_size;
    const float* x    = (const float*)d_in[0];   // [4,2048,1024]
    const float* W    = (const float*)d_in[1];   // [2048,1024]
    const float* bias = (const float*)d_in[2];   // [2048]
    float* out = (float*)d_out;                  // [4,2048,1024]

    float* Qws = (float*)d_ws;                                   // 32 MB
    float* Kws = Qws + (size_t)B_ * H_ * S_ * DH_;               // 32 MB

    // Projection: 32-row tiles, 4 waves/block (32x256 block tile).
    dim3 g1((B_ * S_) / 32, (2 * D_) / 256);                     // (256, 8)
    qk_proj_kernel<<<g1, dim3(128), 0, stream>>>(x, W, bias, Qws, Kws);

    // Attention: 4 waves per 16-query strip; padded LDS score strip.
    const size_t lds_bytes =
        (size_t)(16 * PSTR + 16 + 16 * 8 + 16 * DH_) * sizeof(float);
    dim3 g2(B_ * H_, S_ / 16);                                   // (64, 128)
    attn_kernel<<<g2, dim3(128), lds_bytes, stream>>>(x, Qws, Kws, out);
}